// AnchorKNNEncoder_21629455303111
// MI455X (gfx1250) — compile-verified
//
#include <hip/hip_runtime.h>
#include <hip/hip_bf16.h>

#define K_NEI 4
#define TAU   0.3f
#define EMB   64
#define NB    4096            // B in reference
#define NM    8192            // M in reference
#define NPAIR (2 * NB)        // 8192 (b,node) pairs

typedef __attribute__((ext_vector_type(2))) float v2f;
typedef __attribute__((ext_vector_type(4))) float v4f;
typedef __attribute__((ext_vector_type(8))) float v8f;

__device__ __forceinline__ float gelu_exact(float x) {
    // approximate=False gelu: 0.5*x*(1+erf(x/sqrt(2)))
    return 0.5f * x * (1.0f + erff(x * 0.70710678118654752440f));
}

__device__ __forceinline__ void insert4(unsigned long long k,
                                        unsigned long long& t0, unsigned long long& t1,
                                        unsigned long long& t2, unsigned long long& t3) {
    if (k < t3) {
        t3 = k;
        unsigned long long s;
        if (t3 < t2) { s = t2; t2 = t3; t3 = s; }
        if (t2 < t1) { s = t1; t1 = t2; t2 = s; }
        if (t1 < t0) { s = t0; t0 = t1; t1 = s; }
    }
}

// ---------------------------------------------------------------------------
// Kernel 1: streaming top-4 nearest anchors per (b,node) pair. One wave / pair.
// Memory-bound: 512MB streamed once -> non-temporal b128 loads.
// Writes per pair: topA[4][2] coords + softmax weights[4] into workspace.
// ---------------------------------------------------------------------------
__global__ __launch_bounds__(256) void knn_topk_kernel(
    const float* __restrict__ nodes,  // (B,2,2)
    const float* __restrict__ ancS,   // (B,M,2)
    const float* __restrict__ ancL,   // (B,M,2)
    float* __restrict__ wsA,          // (NPAIR,4,2)
    float* __restrict__ wsW)          // (NPAIR,4)
{
    const int lane = threadIdx.x & 31;
    const int wave = threadIdx.x >> 5;
    const int pair = blockIdx.x * 8 + wave;       // [0, NPAIR)
    const int b    = pair >> 1;
    const int node = pair & 1;

    const float* __restrict__ anc = (node ? ancL : ancS) + (size_t)b * (NM * 2);
    const float qx = nodes[b * 4 + node * 2 + 0];
    const float qy = nodes[b * 4 + node * 2 + 1];

    unsigned long long t0 = ~0ULL, t1 = ~0ULL, t2 = ~0ULL, t3 = ~0ULL;

    // Each iter: wave covers 64 anchors; lane loads 2 anchors (16B, NT).
#pragma unroll 4
    for (int it = 0; it < NM / 64; ++it) {
        const int idx = it * 64 + lane * 2;
        v4f v = __builtin_nontemporal_load((const v4f*)(anc + (size_t)idx * 2));
        float dx0 = v.x - qx, dy0 = v.y - qy;
        float dx1 = v.z - qx, dy1 = v.w - qy;
        float d20 = dx0 * dx0 + dy0 * dy0;
        float d21 = dx1 * dx1 + dy1 * dy1;
        unsigned long long k0 =
            ((unsigned long long)__float_as_uint(d20) << 32) | (unsigned)idx;
        unsigned long long k1 =
            ((unsigned long long)__float_as_uint(d21) << 32) | (unsigned)(idx + 1);
        insert4(k0, t0, t1, t2, t3);
        insert4(k1, t0, t1, t2, t3);
    }

    // 4 rounds: global min over 32 lanes' heads (keys unique: idx in low bits).
    unsigned long long win[4];
#pragma unroll
    for (int r = 0; r < 4; ++r) {
        unsigned long long best = t0;
#pragma unroll
        for (int off = 16; off >= 1; off >>= 1) {
            unsigned long long o = __shfl_xor(best, off, 32);
            if (o < best) best = o;
        }
        if (t0 == best) { t0 = t1; t1 = t2; t2 = t3; t3 = ~0ULL; }
        win[r] = best;
    }

    // softmax(vals / -tau) with vals = -d2  ==  softmax(d2 / tau); d2 ascending.
    float s[4];
#pragma unroll
    for (int r = 0; r < 4; ++r)
        s[r] = __uint_as_float((unsigned)(win[r] >> 32)) * (1.0f / TAU);
    const float mx  = s[3];
    const float sum = expf(s[0] - mx) + expf(s[1] - mx) + expf(s[2] - mx) + expf(s[3] - mx);

    if (lane < 4) {
        const unsigned idx = (unsigned)win[lane];
        wsA[(size_t)pair * 8 + lane * 2 + 0] = anc[(size_t)idx * 2 + 0];
        wsA[(size_t)pair * 8 + lane * 2 + 1] = anc[(size_t)idx * 2 + 1];
        wsW[(size_t)pair * 4 + lane]         = expf(s[lane] - mx) / sum;
    }
}

// ---------------------------------------------------------------------------
// Kernel 2: MLP (2->64 gelu ->64 gelu) + weighted sum, one wave per 16 rows
// (= 4 pairs). Layer 1 computed directly in WMMA A-matrix layout (K=2 FMAs),
// layer 2 via 64x V_WMMA_F32_16X16X4_F32 (fp32 WMMA), bias+gelu in D layout,
// then per-pair softmax-weighted reduction using the D row/lane-half mapping.
// ---------------------------------------------------------------------------
__global__ __launch_bounds__(256) void mlp_wmma_kernel(
    const float* __restrict__ wsA,   // (NPAIR,4,2) rows = pair*4+k
    const float* __restrict__ wsW,   // (NPAIR,4)
    const float* __restrict__ W1,    // (64,2)
    const float* __restrict__ b1,    // (64)
    const float* __restrict__ W2,    // (64,64)
    const float* __restrict__ b2,    // (64)
    float* __restrict__ out)         // hs (B,64) then hl (B,64)
{
    const int lane = threadIdx.x & 31;
    const int wave = threadIdx.x >> 5;
    const int tile = blockIdx.x * 8 + wave;   // [0, NPAIR*4/16) = [0, 2048)
    const int lo   = lane & 15;
    const int hi   = lane >> 4;

    // Row handled by this lane in A-layout: rows 0..15 of the tile.
    const int row = tile * 16 + lo;           // global MLP row = pair*4 + k
    const float x = wsA[(size_t)row * 2 + 0];
    const float y = wsA[(size_t)row * 2 + 1];

    // Layer 1 in WMMA A layout: chunk c covers K = 4c..4c+3 of h;
    // lanes 0-15 hold k = 4c,4c+1; lanes 16-31 hold k = 4c+2,4c+3.
    float a[32];
#pragma unroll
    for (int c = 0; c < 16; ++c) {
        const int e0 = 4 * c + 2 * hi;
        a[2 * c + 0] = gelu_exact(x * W1[e0 * 2 + 0] + y * W1[e0 * 2 + 1] + b1[e0]);
        a[2 * c + 1] = gelu_exact(x * W1[e0 * 2 + 2] + y * W1[e0 * 2 + 3] + b1[e0 + 1]);
    }

    // D rows 0-7 live in lanes 0-15, rows 8-15 in lanes 16-31:
    // this lane-half owns pairs (tile*4 + 2*hi) and (tile*4 + 2*hi + 1).
    const int pairA = tile * 4 + 2 * hi;      // always even -> node 0 (hs)
    float wA[4], wB[4];
#pragma unroll
    for (int r = 0; r < 4; ++r) {
        wA[r] = wsW[(size_t)pairA * 4 + r];
        wB[r] = wsW[(size_t)(pairA + 1) * 4 + r];
    }

#pragma unroll
    for (int t = 0; t < 4; ++t) {             // 16-column N tiles of the 64 outputs
        v8f acc = {0.f, 0.f, 0.f, 0.f, 0.f, 0.f, 0.f, 0.f};
#pragma unroll
        for (int c = 0; c < 16; ++c) {        // K chunks of 4
            // B chunk (4x16): B[k][n] = W2[16t+n][4c+k]; lane n=lo, k = 2*hi,2*hi+1
            const float* bp = W2 + (size_t)(t * 16 + lo) * 64 + 4 * c + 2 * hi;
            v2f bb; bb.x = bp[0]; bb.y = bp[1];
            v2f aa; aa.x = a[2 * c]; aa.y = a[2 * c + 1];
            acc = __builtin_amdgcn_wmma_f32_16x16x4_f32(
                false, aa, false, bb, (short)0, acc, false, false);
        }
        const int col  = t * 16 + lo;
        const float bc = b2[col];
        float o0 = 0.f, o1 = 0.f;
#pragma unroll
        for (int r = 0; r < 4; ++r) {
            o0 += wA[r] * gelu_exact(acc[r] + bc);       // rows 4p..4p+3 of pairA
            o1 += wB[r] * gelu_exact(acc[4 + r] + bc);   // rows of pairA+1
        }
        const int b0 = pairA >> 1;                       // pairA: node 0, pairA+1: node 1
        out[(size_t)0 * NB * EMB + (size_t)b0 * EMB + col] = o0;
        out[(size_t)1 * NB * EMB + (size_t)b0 * EMB + col] = o1;
    }
}

extern "C" void kernel_launch(void* const* d_in, const int* in_sizes, int n_in,
                              void* d_out, int out_size, void* d_ws, size_t ws_size,
                              hipStream_t stream) {
    const float* nodes = (const float*)d_in[0];  // (B,2,2)
    const float* ancS  = (const float*)d_in[1];  // (B,M,2)
    const float* ancL  = (const float*)d_in[2];  // (B,M,2)
    const float* W1    = (const float*)d_in[3];  // (64,2)
    const float* b1    = (const float*)d_in[4];  // (64)
    const float* W2    = (const float*)d_in[5];  // (64,64)
    const float* b2    = (const float*)d_in[6];  // (64)
    float* out = (float*)d_out;

    float* wsA = (float*)d_ws;                   // NPAIR*8 floats (256 KB)
    float* wsW = wsA + (size_t)NPAIR * 8;        // NPAIR*4 floats (128 KB)

    // Kernel 1: 8192 pairs, 1 wave each, 8 waves/block -> 1024 blocks.
    knn_topk_kernel<<<NPAIR / 8, 256, 0, stream>>>(nodes, ancS, ancL, wsA, wsW);

    // Kernel 2: 2048 row-tiles of 16, 1 wave each, 8 waves/block -> 256 blocks.
    mlp_wmma_kernel<<<(NPAIR * 4 / 16) / 8, 256, 0, stream>>>(
        wsA, wsW, W1, b1, W2, b2, out);
}